// ForwardAttention_26319559590458
// MI455X (gfx1250) — compile-verified
//
#include <hip/hip_runtime.h>
#include <hip/hip_bf16.h>

// ---------------- problem constants (from the reference) ----------------
#define B_    64
#define T_    2000
#define DEN_  512
#define DQ_   1024
#define DA_   128
#define NF_   32
#define KW_   31
#define PADW_ 15
#define TTILE_ 16           // t-rows per energy block
#define NTT_   125          // 2000 / 16
#define CHUNK_ 100          // t per context chunk
#define NCHUNK_ 20

typedef __attribute__((ext_vector_type(2))) float v2f;
typedef __attribute__((ext_vector_type(4))) float f32x4;
typedef __attribute__((ext_vector_type(8))) float v8f;

// Fast tanh: prefer gfx1250 hardware v_tanh_f32; else v_exp_f32-based rational.
static __device__ __forceinline__ float fast_tanh(float x) {
#if __has_builtin(__builtin_amdgcn_tanhf)
  return __builtin_amdgcn_tanhf(x);
#elif __has_builtin(__builtin_amdgcn_tanh_f32)
  return __builtin_amdgcn_tanh_f32(x);
#else
  const float ax = fabsf(x);
  const float t  = __expf(-2.0f * ax);          // v_exp_f32
  const float r  = (1.0f - t) / (1.0f + t);
  return copysignf(r, x);
#endif
}

// =======================================================================
// Kernel 1: pq = query @ Wq^T   [64x128] = [64x1024] x [1024x128]
// One wave (32 lanes) per 16x16 output tile, WMMA f32 16x16x4, LDS-staged.
// =======================================================================
__global__ __launch_bounds__(32)
void pq_gemm_wmma(const float* __restrict__ query,   // [B, DQ]
                  const float* __restrict__ Wq,      // [DA, DQ]
                  float* __restrict__ pq)            // [B, DA]
{
  const int tile = blockIdx.x;       // 4 m-tiles * 8 n-tiles = 32
  const int tm   = tile >> 3;
  const int tn   = tile & 7;
  const int lane = threadIdx.x;
  const int h    = lane >> 4;        // half
  const int l    = lane & 15;

  __shared__ float As[16 * 64];      // [m][kk]
  __shared__ float Bs[16 * 64];      // [n][kk]   (row of Wq == column of Wq^T)

  v8f acc = {};
  for (int k0 = 0; k0 < DQ_; k0 += 64) {
    // cooperative coalesced tile load: 256 float4 per tile
    for (int i = lane; i < 256; i += 32) {
      const int r = i >> 4, c4 = i & 15;
      *(((f32x4*)As) + i) = *(const f32x4*)(query + (size_t)(tm * 16 + r) * DQ_ + k0 + c4 * 4);
      *(((f32x4*)Bs) + i) = *(const f32x4*)(Wq    + (size_t)(tn * 16 + r) * DQ_ + k0 + c4 * 4);
    }
    __syncthreads();
#pragma unroll
    for (int ks = 0; ks < 16; ++ks) {
      const int k = ks * 4 + 2 * h;  // A layout: lanes 0-15 hold K=0,1; 16-31 hold K=2,3
      v2f a, b;
      a.x = As[l * 64 + k];  a.y = As[l * 64 + k + 1];
      b.x = Bs[l * 64 + k];  b.y = Bs[l * 64 + k + 1];
      acc = __builtin_amdgcn_wmma_f32_16x16x4_f32(false, a, false, b,
                                                  (short)0, acc, false, false);
    }
    __syncthreads();
  }
  // D layout: VGPR v -> row (8h+v), lane low nibble -> col
#pragma unroll
  for (int vv = 0; vv < 8; ++vv)
    pq[(size_t)(tm * 16 + 8 * h + vv) * DA_ + tn * 16 + l] = acc[vv];
}

// =======================================================================
// Kernel 2: energies[b, t0..t0+15]
//   conv(prev_w,cum_w) -> loc (LDS, A-operand layout)
//   pa = loc^T @ Wld^T via WMMA f32 16x16x4 (16x32x128 tile GEMM)
//   e = sum_a v[a] * tanh(pq + pa + pm)   (pm streamed non-temporal)
// One wave per (t-tile, b).
// =======================================================================
__global__ __launch_bounds__(32)
void energy_kernel(const float* __restrict__ prev_w,  // [B,T]
                   const float* __restrict__ cum_w,   // [B,T]
                   const float* __restrict__ Wconv,   // [NF,2,K]
                   const float* __restrict__ Wld,     // [DA,NF]
                   const float* __restrict__ pq,      // [B,DA]
                   const float* __restrict__ pm,      // [B,T,DA]
                   const float* __restrict__ vvec,    // [DA]
                   const unsigned char* __restrict__ mask, // [B,T]
                   float* __restrict__ energies)      // [B,T]
{
  const int t0   = blockIdx.x * TTILE_;
  const int b    = blockIdx.y;
  const int lane = threadIdx.x;
  const int h    = lane >> 4;
  const int l    = lane & 15;

  __shared__ float wconv_s[NF_ * 2 * KW_];  // 1984
  __shared__ float winp[46], winc[46];      // 16 + 30 halo
  __shared__ float loc_s[NF_ * TTILE_];     // [f*16 + tt]  (== A[k][m], k-major)
  __shared__ float pa_s[TTILE_ * DA_];      // [tt][a]
  __shared__ float pq_s[DA_], v_s[DA_];

  for (int i = lane; i < NF_ * 2 * KW_; i += 32) wconv_s[i] = Wconv[i];
  for (int i = lane; i < DA_; i += 32) {
    pq_s[i] = pq[(size_t)b * DA_ + i];
    v_s[i]  = vvec[i];
  }
  for (int j = lane; j < 46; j += 32) {
    const int t = t0 - PADW_ + j;
    const bool in = (t >= 0) && (t < T_);
    winp[j] = in ? prev_w[(size_t)b * T_ + t] : 0.f;
    winc[j] = in ? cum_w [(size_t)b * T_ + t] : 0.f;
  }
  __syncthreads();

  // --- conv: one filter per lane (cross-correlation, as lax.conv) ---
  {
    const int f = lane;
    const float* w0 = &wconv_s[f * 2 * KW_];
    const float* w1 = w0 + KW_;
#pragma unroll 4
    for (int tt = 0; tt < TTILE_; ++tt) {
      float s = 0.f;
#pragma unroll
      for (int k = 0; k < KW_; ++k)
        s += winp[tt + k] * w0[k] + winc[tt + k] * w1[k];
      loc_s[f * TTILE_ + tt] = s;
    }
  }
  __syncthreads();

  // --- pa[tt][a] = sum_f loc[f][tt] * Wld[a][f] : WMMA M=16,N=16,K=4 x (8 n-tiles, 8 k-steps)
  for (int tn = 0; tn < 8; ++tn) {
    v8f acc = {};
#pragma unroll
    for (int ks = 0; ks < 8; ++ks) {
      const int k = ks * 4 + 2 * h;
      v2f a, bf;
      a.x = loc_s[(k    ) * TTILE_ + l];
      a.y = loc_s[(k + 1) * TTILE_ + l];
      const float* wl = Wld + (size_t)(tn * 16 + l) * NF_ + k;  // B[k][n] = Wld[n][k]
      bf.x = wl[0];  bf.y = wl[1];
      acc = __builtin_amdgcn_wmma_f32_16x16x4_f32(false, a, false, bf,
                                                  (short)0, acc, false, false);
    }
#pragma unroll
    for (int vv = 0; vv < 8; ++vv)
      pa_s[(8 * h + vv) * DA_ + tn * 16 + l] = acc[vv];
  }
  __syncthreads();

  // --- energy: 2 lanes per t-row, each covers 64 of the 128 'a' channels ---
  const int row  = lane >> 1;
  const int part = lane & 1;
  const int t    = t0 + row;
  const float* pmrow = pm + ((size_t)b * T_ + t) * DA_ + part * 64;
  const float* parow = &pa_s[row * DA_ + part * 64];
  const float* pqp   = &pq_s[part * 64];
  const float* vp    = &v_s [part * 64];
  float e = 0.f;
#pragma unroll 4
  for (int i = 0; i < 16; ++i) {
    f32x4 x = __builtin_nontemporal_load(((const f32x4*)pmrow) + i);  // stream pm once
#pragma unroll
    for (int j = 0; j < 4; ++j) {
      const int aa = i * 4 + j;
      e += vp[aa] * fast_tanh(pqp[aa] + parow[aa] + x[j]);
    }
  }
  e += __shfl_xor(e, 1);
  if (part == 0) {
    energies[(size_t)b * T_ + t] = mask[(size_t)b * T_ + t] ? -INFINITY : e;
  }
}

// =======================================================================
// Kernel 3: softmax over T, forward-attention alpha update, renormalize.
// One 256-thread block per batch row; T kept in LDS.
// =======================================================================
__global__ __launch_bounds__(256)
void softmax_alpha_kernel(const float* __restrict__ energies, // [B,T]
                          const float* __restrict__ alpha,    // [B,T]
                          const float* __restrict__ u,        // [B,1]
                          float* __restrict__ alphan)         // [B,T] out
{
  const int b   = blockIdx.x;
  const int tid = threadIdx.x;
  __shared__ float s[T_];
  __shared__ float red[256];

  // max
  float m = -INFINITY;
  for (int t = tid; t < T_; t += 256) {
    const float x = energies[(size_t)b * T_ + t];
    s[t] = x;
    m = fmaxf(m, x);
  }
  red[tid] = m; __syncthreads();
  for (int o = 128; o > 0; o >>= 1) { if (tid < o) red[tid] = fmaxf(red[tid], red[tid + o]); __syncthreads(); }
  m = red[0]; __syncthreads();

  // exp + sum
  float lsum = 0.f;
  for (int t = tid; t < T_; t += 256) {
    const float p = __expf(s[t] - m);
    s[t] = p;
    lsum += p;
  }
  red[tid] = lsum; __syncthreads();
  for (int o = 128; o > 0; o >>= 1) { if (tid < o) red[tid] += red[tid + o]; __syncthreads(); }
  const float inv = 1.f / red[0];
  __syncthreads();

  // forward-attention alpha (training mode) + sum
  const float ub = u[b];
  float nsum = 0.f;
  for (int t = tid; t < T_; t += 256) {
    const float aw = s[t] * inv;
    const float ap = (t > 0) ? alpha[(size_t)b * T_ + t - 1] : 0.f;
    const float na = ((1.f - ub) * alpha[(size_t)b * T_ + t] + ub * ap + 1e-8f) * aw;
    s[t] = na;
    nsum += na;
  }
  __syncthreads();
  red[tid] = nsum; __syncthreads();
  for (int o = 128; o > 0; o >>= 1) { if (tid < o) red[tid] += red[tid + o]; __syncthreads(); }
  const float ninv = 1.f / red[0];
  __syncthreads();

  for (int t = tid; t < T_; t += 256)
    alphan[(size_t)b * T_ + t] = s[t] * ninv;
}

// =======================================================================
// Kernel 4: zero the (poisoned) output.
// =======================================================================
__global__ void zero_kernel(float* __restrict__ out, int n) {
  const int i = blockIdx.x * blockDim.x + threadIdx.x;
  if (i < n) out[i] = 0.f;
}

// =======================================================================
// Kernel 5: context[b,d] = sum_t alphan[b,t] * memory[b,t,d]
// The 256 MB streaming read: grid (20 t-chunks x 64 b), 128 threads, each
// thread owns 4 consecutive d's -> b128 NT loads (512B / wave / instr).
// f32 atomics combine the t-chunks.
// =======================================================================
__global__ __launch_bounds__(128)
void ctx_kernel(const float* __restrict__ alphan,  // [B,T]
                const float* __restrict__ memory,  // [B,T,DEN]
                float* __restrict__ out)           // [B,DEN]
{
  const int chunk = blockIdx.x;
  const int b     = blockIdx.y;
  const int tid   = threadIdx.x;       // 0..127 -> d = tid*4 .. tid*4+3
  const int tbase = chunk * CHUNK_;

  __shared__ float a_s[CHUNK_];
  for (int i = tid; i < CHUNK_; i += 128)
    a_s[i] = alphan[(size_t)b * T_ + tbase + i];
  __syncthreads();

  f32x4 acc = {0.f, 0.f, 0.f, 0.f};
  const f32x4* mrow = (const f32x4*)(memory + ((size_t)b * T_ + tbase) * DEN_) + tid;
#pragma unroll 4
  for (int i = 0; i < CHUNK_; ++i) {
    const f32x4 x = __builtin_nontemporal_load(mrow + (size_t)i * (DEN_ / 4));
    const float a = a_s[i];
    acc.x += a * x.x;  acc.y += a * x.y;
    acc.z += a * x.z;  acc.w += a * x.w;
  }

  float* o = out + (size_t)b * DEN_ + tid * 4;
  atomicAdd(o + 0, acc.x);
  atomicAdd(o + 1, acc.y);
  atomicAdd(o + 2, acc.z);
  atomicAdd(o + 3, acc.w);
}

// =======================================================================
// Launcher
// =======================================================================
extern "C" void kernel_launch(void* const* d_in, const int* in_sizes, int n_in,
                              void* d_out, int out_size, void* d_ws, size_t ws_size,
                              hipStream_t stream) {
  const float* query  = (const float*)d_in[0];   // [B,DQ]
  const float* memory = (const float*)d_in[1];   // [B,T,DEN]
  const float* pm     = (const float*)d_in[2];   // [B,T,DA]
  const float* prev_w = (const float*)d_in[3];   // [B,T]
  const float* cum_w  = (const float*)d_in[4];   // [B,T]
  const float* alpha  = (const float*)d_in[5];   // [B,T]
  const float* u      = (const float*)d_in[6];   // [B,1]
  const float* Wq     = (const float*)d_in[7];   // [DA,DQ]
  const float* Wconv  = (const float*)d_in[8];   // [NF,2,K]
  const float* Wld    = (const float*)d_in[9];   // [DA,NF]
  const float* v      = (const float*)d_in[10];  // [1,DA]
  // d_in[11] Wta, d_in[12] bta: reference discards new_u -> unused
  const unsigned char* mask = (const unsigned char*)d_in[13]; // [B,T] bool

  float* ws       = (float*)d_ws;
  float* pq       = ws;                          // B*DA      = 8192
  float* energies = pq + (size_t)B_ * DA_;       // B*T       = 128000
  float* alphan   = energies + (size_t)B_ * T_;  // B*T       = 128000
  float* out      = (float*)d_out;               // B*DEN     = 32768

  pq_gemm_wmma<<<dim3(32), dim3(32), 0, stream>>>(query, Wq, pq);
  energy_kernel<<<dim3(NTT_, B_), dim3(32), 0, stream>>>(prev_w, cum_w, Wconv, Wld,
                                                         pq, pm, v, mask, energies);
  softmax_alpha_kernel<<<dim3(B_), dim3(256), 0, stream>>>(energies, alpha, u, alphan);
  zero_kernel<<<dim3((B_ * DEN_ + 255) / 256), dim3(256), 0, stream>>>(out, B_ * DEN_);
  ctx_kernel<<<dim3(NCHUNK_, B_), dim3(128), 0, stream>>>(alphan, memory, out);
}